// LightGCN_1486058684584
// MI455X (gfx1250) — compile-verified
//
#include <hip/hip_runtime.h>
#include <hip/hip_bf16.h>

#define N_USERS 100000
#define N_ITEMS 50000
#define N_NODES (N_USERS + N_ITEMS)   // 150000
#define DIM 64
#define LAYERS 3
#define BATCH 16384

typedef __attribute__((ext_vector_type(2))) float v2f;
typedef __attribute__((ext_vector_type(8))) float v8f;

// ---------------------------------------------------------------------------
// init: A = C = concat(user_emb, item_emb); B = 0.   float4-vectorized.
// ---------------------------------------------------------------------------
__global__ void lgcn_init(const float4* __restrict__ ue, const float4* __restrict__ ie,
                          float4* __restrict__ A, float4* __restrict__ B,
                          float4* __restrict__ C) {
    const int n4u = N_USERS * (DIM / 4);            // 1,600,000
    const int n4  = N_NODES * (DIM / 4);            // 2,400,000
    int i = blockIdx.x * blockDim.x + threadIdx.x;
    if (i >= n4) return;
    float4 v = (i < n4u) ? ue[i] : ie[i - n4u];
    A[i] = v;
    C[i] = v;
    B[i] = make_float4(0.f, 0.f, 0.f, 0.f);
}

// ---------------------------------------------------------------------------
// spmm: one wave per edge. B[row] += val * A[col]. Lane d handles dims 2d,2d+1
// via a float2 load and two native global_atomic_add_f32.
// Edge scalars are wave-uniform (readfirstlane) -> scalar loads.
// ---------------------------------------------------------------------------
__global__ void lgcn_spmm(const float* __restrict__ src, float* __restrict__ dst,
                          const float* __restrict__ ev, const int* __restrict__ er,
                          const int* __restrict__ ec, int nE) {
    const int wavesPerBlock = blockDim.x >> 5;
    int e = __builtin_amdgcn_readfirstlane(blockIdx.x * wavesPerBlock + (threadIdx.x >> 5));
    if (e >= nE) return;
    const int lane = threadIdx.x & 31;

    const float val = ev[e];
    const int   row = er[e];
    const int   col = ec[e];

    const v2f m = *(const v2f*)(src + (size_t)col * DIM + lane * 2);
    float* d = dst + (size_t)row * DIM + lane * 2;
    unsafeAtomicAdd(d + 0, val * m[0]);
    unsafeAtomicAdd(d + 1, val * m[1]);
}

// ---------------------------------------------------------------------------
// combine: C += B; A = B; B = 0  (prepares next layer, deterministic state).
// ---------------------------------------------------------------------------
__global__ void lgcn_combine(float4* __restrict__ A, float4* __restrict__ B,
                             float4* __restrict__ C) {
    const int n4 = N_NODES * (DIM / 4);
    int i = blockIdx.x * blockDim.x + threadIdx.x;
    if (i >= n4) return;
    float4 b = B[i];
    float4 c = C[i];
    c.x += b.x; c.y += b.y; c.z += b.z; c.w += b.w;
    C[i] = c;
    A[i] = b;
    B[i] = make_float4(0.f, 0.f, 0.f, 0.f);
}

// ---------------------------------------------------------------------------
// readout: out[b] = (acc[u_b] . acc[N_USERS + i_b]) / 16  via WMMA f32 16x16x4.
// One wave handles 16 batch elements. A-frag = 16 user-rows x 4 dims,
// B-frag = 4 dims x 16 item-cols; 16 K-steps cover DIM=64. Diagonal of the
// 16x16 f32 accumulator is the 16 dot products.
//
// ISA fragment layout (32-bit, 16x4 A and 4x16 B):
//   lane = 16*h + r (h in {0,1}, r in 0..15):
//     A vgpr j = A[M=r][K = 2h + j],  B vgpr j = B[K = 2h + j][N = r]
// -> both load the same per-lane float2 at row r, dim 4c + 2h.
// C/D layout: lanes 0-15 hold M=vgpr (0..7), lanes 16-31 hold M=vgpr+8; N=lane%16.
// Diagonal element n sits at vgpr (n&7) on lane n (n<8) or lane n+16 (n>=8).
// ---------------------------------------------------------------------------
__global__ void lgcn_dot_wmma(const float* __restrict__ C, const int* __restrict__ users,
                              const int* __restrict__ items, float* __restrict__ out) {
    const int wave = blockIdx.x * (blockDim.x >> 5) + (threadIdx.x >> 5); // 0..1023
    const int lane = threadIdx.x & 31;
    const int lo   = lane & 15;       // row / col index within tile
    const int hi   = lane >> 4;       // K-half selector

    const int b = wave * 16 + lo;     // batch element whose row this lane feeds
    const float* __restrict__ urow = C + (size_t)users[b] * DIM;
    const float* __restrict__ irow = C + (size_t)(N_USERS + items[b]) * DIM;

    v8f acc = {};
#pragma unroll
    for (int c = 0; c < DIM / 4; ++c) {
        const int k = c * 4 + hi * 2;
        v2f a  = *(const v2f*)(urow + k);
        v2f bb = *(const v2f*)(irow + k);
        acc = __builtin_amdgcn_wmma_f32_16x16x4_f32(
            /*neg_a=*/false, a, /*neg_b=*/false, bb,
            /*c_mod=*/(short)0, acc, /*reuse_a=*/false, /*reuse_b=*/false);
    }

    // extract diagonal
    const int v = lo & 7;
    float d = acc[0];
#pragma unroll
    for (int j = 1; j < 8; ++j)
        if (v == j) d = acc[j];

    const bool valid = (hi == 0) ? (lo < 8) : (lo >= 8);
    if (valid)
        out[wave * 16 + lo] = d * 0.0625f;   // (acc/4).(acc/4) => /16
}

// ---------------------------------------------------------------------------
extern "C" void kernel_launch(void* const* d_in, const int* in_sizes, int n_in,
                              void* d_out, int out_size, void* d_ws, size_t ws_size,
                              hipStream_t stream) {
    const float* user_emb = (const float*)d_in[0];
    const float* item_emb = (const float*)d_in[1];
    const float* edge_val = (const float*)d_in[2];
    const int*   edge_row = (const int*)d_in[3];
    const int*   edge_col = (const int*)d_in[4];
    const int*   users    = (const int*)d_in[5];
    const int*   items    = (const int*)d_in[6];
    const int    nE       = in_sizes[2];
    float*       out      = (float*)d_out;

    const size_t tbl = (size_t)N_NODES * DIM;        // 9.6M floats
    float* A = (float*)d_ws;                          // current layer emb
    float* B = A + tbl;                               // next layer emb (atomic target)
    float* C = B + tbl;                               // running sum over layers

    const int n4 = N_NODES * (DIM / 4);              // 2,400,000 float4s
    const int initBlocks = (n4 + 255) / 256;

    lgcn_init<<<initBlocks, 256, 0, stream>>>(
        (const float4*)user_emb, (const float4*)item_emb,
        (float4*)A, (float4*)B, (float4*)C);

    const int spmmBlocks = (nE + 7) / 8;             // 8 waves of 32 per block
    for (int l = 0; l < LAYERS; ++l) {
        lgcn_spmm<<<spmmBlocks, 256, 0, stream>>>(A, B, edge_val, edge_row, edge_col, nE);
        lgcn_combine<<<initBlocks, 256, 0, stream>>>((float4*)A, (float4*)B, (float4*)C);
    }

    // 16384 batch / 16 per wave = 1024 waves; 4 waves per block -> 256 blocks
    lgcn_dot_wmma<<<256, 128, 0, stream>>>(C, users, items, out);
}